// CartesianMACE_33036888441241
// MI455X (gfx1250) — compile-verified
//
#include <hip/hip_runtime.h>
#include <math.h>

#define N_NODES 20000
#define N_EDGES 320000
#define CH      16
#define NLAYERS 2
#define NGRAPH  64
#define NOUT    2
#define NCOMP   13          // 1 + 3 + 9
#define HDIM    208         // NCOMP * CH
#define RANKS   3

typedef __attribute__((ext_vector_type(2))) float v2f;
typedef __attribute__((ext_vector_type(8))) float v8f;

// ---------------------------------------------------------------- utilities
__global__ void zero_kernel(float* __restrict__ p, int n) {
  int i = blockIdx.x * blockDim.x + threadIdx.x;
  if (i < n) p[i] = 0.0f;
}

// ------------------------------------------------- per-edge geometry + radial
__global__ void geom_kernel(const float* __restrict__ pos,
                            const int* __restrict__ snd,
                            const int* __restrict__ rcv,
                            float* __restrict__ u_out,
                            float* __restrict__ rad_out) {
  int e = blockIdx.x * blockDim.x + threadIdx.x;
  if (e >= N_EDGES) return;
  int s = snd[e], r = rcv[e];
  float rx = pos[s * 3 + 0] - pos[r * 3 + 0];
  float ry = pos[s * 3 + 1] - pos[r * 3 + 1];
  float rz = pos[s * 3 + 2] - pos[r * 3 + 2];
  float d  = sqrtf(rx * rx + ry * ry + rz * rz);
  float inv = 1.0f / d;
  u_out[e * 3 + 0] = rx * inv;
  u_out[e * 3 + 1] = ry * inv;
  u_out[e * 3 + 2] = rz * inv;
  const float PI = 3.14159265358979323846f;
#pragma unroll
  for (int c = 0; c < CH; ++c)
    rad_out[e * CH + c] = __sinf((float)(c + 1) * PI * d) * inv;
}

// ---------------------------------------------------------------- init h
// layout: h[n][comp][ch], comp 0 = rank0, 1..3 = rank1 (k), 4..12 = rank2 (i*3+j)
__global__ void init_h_kernel(const float* __restrict__ emb,
                              const int* __restrict__ atoms,
                              float* __restrict__ h) {
  int idx = blockIdx.x * blockDim.x + threadIdx.x;
  if (idx >= N_NODES * HDIM) return;
  int n = idx / HDIM, k = idx % HDIM;
  h[idx] = (k < CH) ? emb[atoms[n] * CH + k] : 0.0f;
}

// -------------------------------------------------- edge messages via WMMA
// One wave = 16 edges. For each of the 13 components, one 16x16x16 GEMM
// D(out_ch x edges) = ab_w[rank](16x16) * msg(in_ch x edges), built from four
// chained V_WMMA_F32_16X16X4_F32, then atomic scatter-add into a_buf[rcv].
__global__ __launch_bounds__(128)
void edge_msg_kernel(const float* __restrict__ h,
                     const float* __restrict__ u_in,
                     const float* __restrict__ rad_in,
                     const int* __restrict__ snd,
                     const int* __restrict__ rcv,
                     const float* __restrict__ abw,   // (RANKS,16,16) this layer
                     float* __restrict__ a_buf,
                     int layer) {
  __shared__ float msg[4 * 16 * HDIM];          // 4 waves * 16 edges * 208
  const int lane = threadIdx.x & 31;
  const int wave = threadIdx.x >> 5;
  const int ebase = (blockIdx.x * 4 + wave) * 16;
  float* m = &msg[wave * 16 * HDIM];

  const int colHalf = lane >> 4;   // 0: lanes 0-15, 1: lanes 16-31
  const int ln      = lane & 15;

  // ---- preload A (ab_w) in the 16x4 f32 A layout, all 3 ranks, 4 K-steps
  float A0[RANKS][4], A1[RANKS][4];
#pragma unroll
  for (int r = 0; r < RANKS; ++r)
#pragma unroll
    for (int s = 0; s < 4; ++s) {
      int k = 4 * s + 2 * colHalf;
      A0[r][s] = abw[r * 256 + ln * 16 + k];
      A1[r][s] = abw[r * 256 + ln * 16 + k + 1];
    }

  // ---- build message tensor in LDS: msg[e][comp][j]
  for (int el = 0; el < 16; ++el) {
    int ge = ebase + el;
    int sn = snd[ge];
    float ux = u_in[ge * 3 + 0], uy = u_in[ge * 3 + 1], uz = u_in[ge * 3 + 2];
    float T[NCOMP];
    T[0] = 1.0f;
    T[1] = ux;      T[2] = uy;      T[3] = uz;
    T[4] = ux * ux; T[5] = ux * uy; T[6] = ux * uz;
    T[7] = uy * ux; T[8] = uy * uy; T[9] = uy * uz;
    T[10] = uz * ux; T[11] = uz * uy; T[12] = uz * uz;
    const float* hrow = h + (size_t)sn * HDIM;
    for (int r = lane; r < HDIM; r += 32) {
      int comp = r >> 4, j = r & 15;
      float rad = rad_in[ge * CH + j];
      float val = hrow[j] * T[comp];          // edge_attr * h0(snd)
      if (layer > 0) val += hrow[r];          // + h[c](snd) * rad
      m[el * HDIM + r] = rad * val;
    }
  }
  __syncthreads();

  const int node = rcv[ebase + ln];

#pragma unroll
  for (int t = 0; t < NCOMP; ++t) {
    const int rk = (t == 0) ? 0 : ((t < 4) ? 1 : 2);
    v8f acc = {0.f, 0.f, 0.f, 0.f, 0.f, 0.f, 0.f, 0.f};
#pragma unroll
    for (int s = 0; s < 4; ++s) {
      v2f av = {A0[rk][s], A1[rk][s]};
      int j = 4 * s + 2 * colHalf;            // B 4x16 layout (symmetric to A)
      v2f bv = {m[ln * HDIM + t * 16 + j], m[ln * HDIM + t * 16 + j + 1]};
      acc = __builtin_amdgcn_wmma_f32_16x16x4_f32(
          false, av, false, bv, (short)0, acc, false, false);
    }
    // D layout: lane -> column (edge), VGPR v -> row (out channel) + 8*colHalf
    float* dst = a_buf + (size_t)node * HDIM + t * 16 + colHalf * 8;
#pragma unroll
    for (int v = 0; v < 8; ++v) atomicAdd(&dst[v], acc[v]);
  }
}

// ---------------------------------------------------------------- node update
__global__ void node_update_kernel(const float* __restrict__ a_buf,
                                   float* __restrict__ h,
                                   const float* __restrict__ ws,   // (NU,RANKS,CH) this layer
                                   const float* __restrict__ chw,  // (RANKS,16,16) this layer
                                   const float* __restrict__ msw) {
  int idx = blockIdx.x * blockDim.x + threadIdx.x;
  if (idx >= N_NODES * CH) return;
  int n = idx >> 4, ch = idx & 15;
  float a0 = a_buf[(size_t)n * HDIM + ch];

  float fac[RANKS], cw[RANKS], mw[RANKS];
#pragma unroll
  for (int r = 0; r < RANKS; ++r) {
    fac[r] = ws[r * CH + ch] + ws[RANKS * CH + r * CH + ch] * a0; // nu=0 + nu=1*a0
    float c = 0.f, mm = 0.f;
#pragma unroll
    for (int i = 0; i < 16; ++i) {
      c  += chw[r * 256 + i * 16 + ch];
      mm += msw[r * 256 + i * 16 + ch];
    }
    cw[r] = c; mw[r] = mm;
  }
#pragma unroll
  for (int comp = 0; comp < NCOMP; ++comp) {
    int r = (comp == 0) ? 0 : ((comp < 4) ? 1 : 2);
    size_t o = (size_t)n * HDIM + comp * 16 + ch;
    float mval = fac[r] * a_buf[o];
    h[o] = h[o] * cw[r] + mval * mw[r];
  }
}

// ---------------------------------------------------------------- pooling
__global__ void pool_kernel(const float* __restrict__ h,
                            const int* __restrict__ batch,
                            float* __restrict__ pooled) {
  int n = blockIdx.x;
  int k = threadIdx.x;
  if (k < HDIM)
    atomicAdd(&pooled[batch[n] * HDIM + k], h[(size_t)n * HDIM + k]);
}

// ---------------------------------------------------------------- predictor
__global__ void predict_kernel(const float* __restrict__ pooled,
                               const float* __restrict__ W,
                               const float* __restrict__ b,
                               float* __restrict__ out) {
  int tid = threadIdx.x;
  if (tid >= NGRAPH * NOUT) return;
  int g = tid / NOUT, o = tid % NOUT;
  float s = b[o];
  for (int f = 0; f < HDIM; ++f) {
    int my;
    if (f < 16)        { my = f; }                                 // h0: ch
    else if (f < 64)   { int t = f - 16; my = (1 + (t % 3)) * 16 + (t / 3); }  // h1: (ch,k)
    else               { int t = f - 64; my = (4 + (t % 9)) * 16 + (t / 9); }  // h2: (ch,i,j)
    s += pooled[g * HDIM + my] * W[f * NOUT + o];
  }
  out[g * NOUT + o] = s;
}

// ---------------------------------------------------------------- launcher
extern "C" void kernel_launch(void* const* d_in, const int* in_sizes, int n_in,
                              void* d_out, int out_size, void* d_ws, size_t ws_size,
                              hipStream_t stream) {
  const float* pos   = (const float*)d_in[0];
  const float* emb   = (const float*)d_in[1];
  const float* ab_w  = (const float*)d_in[2];   // (L,RANKS,16,16)
  const float* ws_w  = (const float*)d_in[3];   // (L,NU,RANKS,16)
  const float* ch_w  = (const float*)d_in[4];   // (L,RANKS,16,16)
  const float* ms_w  = (const float*)d_in[5];   // (L,RANKS,16,16)
  const float* predW = (const float*)d_in[6];   // (208,2)
  const float* predb = (const float*)d_in[7];   // (2,)
  const int*   atoms = (const int*)d_in[8];
  const int*   eidx  = (const int*)d_in[9];     // (2,E)
  const int*   batch = (const int*)d_in[10];
  float* out = (float*)d_out;
  const int* snd = eidx;
  const int* rcv = eidx + N_EDGES;

  float* ws      = (float*)d_ws;
  float* u_buf   = ws;                                   // 3*E
  float* rad_buf = u_buf + (size_t)3 * N_EDGES;          // 16*E
  float* h_buf   = rad_buf + (size_t)16 * N_EDGES;       // N*208
  float* a_buf   = h_buf + (size_t)N_NODES * HDIM;       // N*208
  float* pooled  = a_buf + (size_t)N_NODES * HDIM;       // 64*208

  geom_kernel<<<(N_EDGES + 255) / 256, 256, 0, stream>>>(pos, snd, rcv, u_buf, rad_buf);
  init_h_kernel<<<(N_NODES * HDIM + 255) / 256, 256, 0, stream>>>(emb, atoms, h_buf);

  for (int l = 0; l < NLAYERS; ++l) {
    zero_kernel<<<(N_NODES * HDIM + 255) / 256, 256, 0, stream>>>(a_buf, N_NODES * HDIM);
    edge_msg_kernel<<<N_EDGES / 64, 128, 0, stream>>>(
        h_buf, u_buf, rad_buf, snd, rcv, ab_w + l * RANKS * 256, a_buf, l);
    node_update_kernel<<<(N_NODES * CH + 255) / 256, 256, 0, stream>>>(
        a_buf, h_buf, ws_w + l * 2 * RANKS * CH, ch_w + l * RANKS * 256,
        ms_w + l * RANKS * 256);
  }

  zero_kernel<<<(NGRAPH * HDIM + 255) / 256, 256, 0, stream>>>(pooled, NGRAPH * HDIM);
  pool_kernel<<<N_NODES, 256, 0, stream>>>(h_buf, batch, pooled);
  predict_kernel<<<1, 128, 0, stream>>>(pooled, predW, predb, out);
}